// DEPTree_21182778704986
// MI455X (gfx1250) — compile-verified
//
#include <hip/hip_runtime.h>
#include <hip/hip_bf16.h>

// ---------------------------------------------------------------------------
// DEP-tree evaluation for MI455X (gfx1250, wave32, WMMA).
//
// Bucket nodes by weight-id on device, then run grouped 16-row GEMM tiles
// through V_WMMA_F32_16X16X4_F32 (fp32 in / fp32 acc, matching the fp32
// reference). Each 256-thread block = 8 waves; wave w computes a 16x16
// output tile (cols [16w,16w+16)) with a fully unrolled K=128 loop split
// into two independent accumulator chains (2 x 16 wmma) for matrix-pipe ILP.
// Invalid tile rows read from a zeroed dummy row (pointer select, no per-k
// masking VALU work). Sibling max-reduction + parent max is a cheap
// elementwise pass. Graph-capture safe (hipMemsetAsync + kernels only).
// ---------------------------------------------------------------------------

typedef __attribute__((ext_vector_type(2))) float v2f;
typedef __attribute__((ext_vector_type(8))) float v8f;

#define TREE_N   21845   // (4^8 - 1) / 3
#define TREE_D   128

// ---- bucketing helpers ----------------------------------------------------

__global__ void count_ids(const int* __restrict__ ids, int n,
                          int* __restrict__ counts) {
    int i = blockIdx.x * blockDim.x + threadIdx.x;
    if (i < n) atomicAdd(&counts[ids[i]], 1);
}

__global__ void scan_bins(const int* __restrict__ counts, int nbins,
                          int* __restrict__ starts, int* __restrict__ cursor) {
    if (blockIdx.x == 0 && threadIdx.x == 0) {
        int run = 0;
        for (int b = 0; b < nbins; ++b) {
            starts[b] = run;
            cursor[b] = run;
            run += counts[b];
        }
    }
}

__global__ void scatter_ids(const int* __restrict__ ids, int n,
                            int* __restrict__ cursor, int* __restrict__ list) {
    int i = blockIdx.x * blockDim.x + threadIdx.x;
    if (i < n) {
        int p = atomicAdd(&cursor[ids[i]], 1);
        list[p] = i;   // row id (global node id for pos phase, local child idx for dep phase)
    }
}

// ---- grouped GEMM: out[row] = relu(in[row] (1x128) * W_b (128x128) + bias_b)
//
// grid.x = bucket (== weight id), grid.y = 16-row tile within bucket.
// Block = 256 threads = 8 waves; wave w handles output columns [16w, 16w+16).
//
// Fragment layouts (32-bit operands, wave32, per CDNA5 ISA):
//   A 16x4 : lanes 0-15 -> M=lane, VGPR0/1 = K=0/1 ; lanes 16-31 -> K=2/3
//   B 4x16 : VGPR0 = row K=0 (lanes 0-15) / K=2 (lanes 16-31); VGPR1 = K=1 / K=3
//   C/D    : VGPR i -> row i (lanes 0-15, N=lane) and row i+8 (lanes 16-31)
// All control flow around the wmma loop is block-uniform => EXEC all-ones.
// ---------------------------------------------------------------------------
__global__ __launch_bounds__(256)
void group_gemm16_relu(const float* __restrict__ in_base,   // [*,128] input rows
                       const int*   __restrict__ list,      // bucket-ordered row ids
                       const int*   __restrict__ starts,    // per-bucket start
                       const int*   __restrict__ counts,    // per-bucket count
                       const float* __restrict__ Wbase,     // [numW,128,128] (d,e)
                       const float* __restrict__ Bbase,     // [numW,128]
                       const float* __restrict__ zrow,      // 128 zeros (dummy row)
                       float*       __restrict__ out_base)  // [*,128] output rows
{
    const int b   = blockIdx.x;
    const int cnt = counts[b];                 // uniform across block
    const int t   = blockIdx.y;
    if (t * 16 >= cnt) return;                 // uniform branch, EXEC stays full

    const int start = starts[b];
    const int lane  = threadIdx.x & 31;
    const int wave  = threadIdx.x >> 5;
    const int colb  = wave << 4;               // column base for this wave
    const int m     = lane & 15;
    const int khalf = (lane >> 4) << 1;        // 0 for lanes 0-15, 2 for lanes 16-31

    // A-row selection: invalid rows read from the zero row (pointer select,
    // v_cndmask once — keeps the inner loop free of masking VALU work).
    const int  ridx  = t * 16 + m;
    const bool valid = ridx < cnt;
    const int  row   = list[start + (valid ? ridx : (cnt - 1))];
    const float* __restrict__ arow =
        valid ? (in_base + (size_t)row * TREE_D) : zrow;
    const float* __restrict__ W = Wbase + (size_t)b * (TREE_D * TREE_D);

    // Warm the weight tile path (global_prefetch_b8); W stays L2-resident anyway.
    __builtin_prefetch(W + (size_t)khalf * TREE_D + colb + m, 0, 1);

    // Two independent accumulator chains (even/odd k-steps) for WMMA ILP.
    v8f acc0 = {};
    v8f acc1 = {};
#pragma unroll
    for (int k = 0; k < TREE_D; k += 8) {
        v2f a0, b0, a1, b1;
        a0.x = arow[k + khalf + 0];
        a0.y = arow[k + khalf + 1];
        b0.x = W[(size_t)(k + khalf + 0) * TREE_D + colb + m];
        b0.y = W[(size_t)(k + khalf + 1) * TREE_D + colb + m];
        a1.x = arow[k + 4 + khalf + 0];
        a1.y = arow[k + 4 + khalf + 1];
        b1.x = W[(size_t)(k + 4 + khalf + 0) * TREE_D + colb + m];
        b1.y = W[(size_t)(k + 4 + khalf + 1) * TREE_D + colb + m];
        // v_wmma_f32_16x16x4_f32: (neg_a, A, neg_b, B, c_mod, C, reuse_a, reuse_b)
        acc0 = __builtin_amdgcn_wmma_f32_16x16x4_f32(false, a0, false, b0,
                                                     (short)0, acc0, false, false);
        acc1 = __builtin_amdgcn_wmma_f32_16x16x4_f32(false, a1, false, b1,
                                                     (short)0, acc1, false, false);
    }

    const float bias = Bbase[(size_t)b * TREE_D + colb + m];
#pragma unroll
    for (int i = 0; i < 8; ++i) {
        const int mo = i + ((lane >> 4) << 3);   // C/D row held by this lane/VGPR
        const int ro = t * 16 + mo;
        if (ro < cnt) {                          // divergence only after all wmma
            const int r = list[start + ro];
            out_base[(size_t)r * TREE_D + colb + m] =
                fmaxf(acc0[i] + acc1[i] + bias, 0.0f);
        }
    }
}

// ---- sibling max(4) + parent combine --------------------------------------
__global__ void max_reduce4(const float* __restrict__ u,      // [n_parent*4,128]
                            const float* __restrict__ x_lvl,  // [n_parent,128]
                            int n_parent,
                            float* __restrict__ z_out)        // [n_parent,128]
{
    int idx = blockIdx.x * blockDim.x + threadIdx.x;
    int total = n_parent * TREE_D;
    if (idx >= total) return;
    int p   = idx >> 7;
    int col = idx & 127;
    const float* up = u + ((size_t)p * 4) * TREE_D + col;
    float mx = up[0];
    mx = fmaxf(mx, up[1 * TREE_D]);
    mx = fmaxf(mx, up[2 * TREE_D]);
    mx = fmaxf(mx, up[3 * TREE_D]);
    z_out[idx] = fmaxf(x_lvl[idx], mx);
}

// ---------------------------------------------------------------------------

extern "C" void kernel_launch(void* const* d_in, const int* in_sizes, int n_in,
                              void* d_out, int out_size, void* d_ws, size_t ws_size,
                              hipStream_t stream) {
    (void)in_sizes; (void)n_in; (void)out_size; (void)ws_size;

    const float* emb     = (const float*)d_in[0];
    const int*   pos_ids = (const int*)  d_in[1];
    const int*   dep_ids = (const int*)  d_in[2];
    const float* pos_W   = (const float*)d_in[3];
    const float* pos_b   = (const float*)d_in[4];
    const float* dep_W   = (const float*)d_in[5];
    const float* dep_b   = (const float*)d_in[6];
    float*       out     = (float*)d_out;

    const int N = TREE_N, D = TREE_D;
    const int NUM_POS = 18, NUM_DEP = 46;
    const int ls[8] = {0, 1, 5, 21, 85, 341, 1365, 5461}; // level_start(l)

    // Workspace carve (~24 MB total).
    char*  ws     = (char*)d_ws;
    float* x      = (float*)ws;                      // N*128
    float* u      = x  + (size_t)N * D;              // 16384*128
    float* zA     = u  + (size_t)16384 * D;          // 4096*128
    float* zB     = zA + (size_t)4096 * D;           // 4096*128
    float* zrow   = zB + (size_t)4096 * D;           // 128 zeros (dummy A row)
    int*   list   = (int*)(zrow + D);                // N ints
    int*   counts = list + ((N + 63) & ~63);         // 64 ints
    int*   starts = counts + 64;                     // 64 ints
    int*   cursor = starts + 64;                     // 64 ints

    hipMemsetAsync(zrow, 0, D * sizeof(float), stream);

    // ---- Phase 1: x[n] = relu(emb[n] @ pos_W[pid] + pos_b[pid]) ----
    hipMemsetAsync(counts, 0, 64 * sizeof(int), stream);
    count_ids  <<<(N + 255) / 256, 256, 0, stream>>>(pos_ids, N, counts);
    scan_bins  <<<1, 32, 0, stream>>>(counts, NUM_POS, starts, cursor);
    scatter_ids<<<(N + 255) / 256, 256, 0, stream>>>(pos_ids, N, cursor, list);
    {
        dim3 g(NUM_POS, (N + 15) / 16);
        group_gemm16_relu<<<g, 256, 0, stream>>>(emb, list, starts, counts,
                                                 pos_W, pos_b, zrow, x);
    }

    // ---- Phase 2: bottom-up tree reduction ----
    const float* z_in = x + (size_t)ls[7] * D;       // leaves: z = x[level 7]
    float* zbufs[2] = { zA, zB };
    int pp = 0;
    for (int l = 6; l >= 0; --l) {
        const int n_c = 1 << (2 * (l + 1));          // 4^(l+1) children
        const int cs  = ls[l + 1];

        hipMemsetAsync(counts, 0, 64 * sizeof(int), stream);
        count_ids  <<<(n_c + 255) / 256, 256, 0, stream>>>(dep_ids + cs, n_c, counts);
        scan_bins  <<<1, 32, 0, stream>>>(counts, NUM_DEP, starts, cursor);
        scatter_ids<<<(n_c + 255) / 256, 256, 0, stream>>>(dep_ids + cs, n_c, cursor, list);

        dim3 g(NUM_DEP, (n_c + 15) / 16);
        group_gemm16_relu<<<g, 256, 0, stream>>>(z_in, list, starts, counts,
                                                 dep_W, dep_b, zrow, u);

        const int n_p = n_c >> 2;
        float* z_out = (l == 0) ? out : zbufs[pp];
        const int total = n_p * D;
        max_reduce4<<<(total + 255) / 256, 256, 0, stream>>>(
            u, x + (size_t)ls[l] * D, n_p, z_out);

        z_in = z_out;
        pp ^= 1;
    }
}